// GCN_9423158247570
// MI455X (gfx1250) — compile-verified
//
#include <hip/hip_runtime.h>
#include <hip/hip_bf16.h>

typedef __attribute__((ext_vector_type(2))) float v2f;
typedef __attribute__((ext_vector_type(8))) float v8f;

// ---------------------------------------------------------------------------
// Degree / normalization kernels
// ---------------------------------------------------------------------------
__global__ void k_deg_init(float* __restrict__ deg, int n) {
    int i = blockIdx.x * blockDim.x + threadIdx.x;
    if (i < n) deg[i] = 1.0f;  // self-loop weight
}

__global__ void k_deg_scatter(const int* __restrict__ dst, const float* __restrict__ ew,
                              float* __restrict__ deg, int E) {
    int e = blockIdx.x * blockDim.x + threadIdx.x;
    if (e < E) atomicAdd(&deg[dst[e]], ew[e]);
}

__global__ void k_rsqrt_inplace(float* __restrict__ d, int n) {
    int i = blockIdx.x * blockDim.x + threadIdx.x;
    if (i < n) d[i] = rsqrtf(d[i]);  // deg >= 1 always (self-loop)
}

// ---------------------------------------------------------------------------
// Pack B[K x ncols] into WMMA-fragment layout, zero-padded to npad columns:
//   P[(k/2)*npad + n] = ( B[k][n], B[k+1][n] )   as float2
// A lane needing K-pair (k+2*half, k+2*half+1) at column n then does ONE
// aligned b64 load -> no divergence, no row-strided scalar loads.
// ---------------------------------------------------------------------------
__global__ void k_pack_b(const float* __restrict__ B, float* __restrict__ P,
                         int K, int ncols, int npad) {
    int idx = blockIdx.x * blockDim.x + threadIdx.x;  // over (K/2)*npad
    if (idx >= (K / 2) * npad) return;
    int kp = idx / npad;
    int n  = idx - kp * npad;
    float v0 = 0.0f, v1 = 0.0f;
    if (n < ncols) {
        v0 = B[(2 * kp) * ncols + n];
        v1 = B[(2 * kp + 1) * ncols + n];
    }
    P[2 * idx]     = v0;
    P[2 * idx + 1] = v1;
}

// ---------------------------------------------------------------------------
// fp32 WMMA GEMM:  C[nrows x (NT*16)] = A[nrows x K] * Bp (packed, padded).
// One 16-row tile per wave. Straight-line per wave: EXEC stays all-ones
// around every V_WMMA_F32_16X16X4_F32 (ISA §7.12 requirement).
// Fragment layout (ISA §7.12.2): lanes 0-15 carry K=k,k+1 ; lanes 16-31
// carry K=k+2,k+3 ; C VGPR r -> row r (lanes 0-15) / row r+8 (lanes 16-31).
// ---------------------------------------------------------------------------
template <int K, int NT>
__global__ void k_gemm_wmma_f32(const float* __restrict__ A, const float* __restrict__ Bp,
                                float* __restrict__ C, int nrows) {
    const int lane = threadIdx.x & 31;
    const int wave = threadIdx.x >> 5;
    const int wavesPerBlock = blockDim.x >> 5;
    const int tile = blockIdx.x * wavesPerBlock + wave;
    const int row_base = tile * 16;
    if (row_base >= nrows) return;  // wave-uniform exit

    const int half = lane >> 4;
    const int l16  = lane & 15;
    const int NPAD = NT * 16;

    v8f acc[NT] = {};

    // A: this lane reads the contiguous pair (k+2*half, k+2*half+1) of its row
    const float* __restrict__ arow = A + (long long)(row_base + l16) * K + 2 * half;
    // Bp: row index = kpair = k/2 + half, column = t*16 + l16
    const float* __restrict__ brow = Bp + 2 * (long long)(half * NPAD + l16);

#pragma unroll 4
    for (int k = 0; k < K; k += 4) {
        v2f a = *(const v2f*)(arow + k);
#pragma unroll
        for (int t = 0; t < NT; ++t) {
            v2f b = *(const v2f*)(brow + 2 * ((long long)(k / 2) * NPAD + t * 16));
            acc[t] = __builtin_amdgcn_wmma_f32_16x16x4_f32(
                false, a, false, b, (short)0, acc[t], false, false);
        }
    }

#pragma unroll
    for (int t = 0; t < NT; ++t) {
        const int col = t * 16 + l16;
#pragma unroll
        for (int r = 0; r < 8; ++r) {
            const int row = row_base + r + 8 * half;
            C[(long long)row * NPAD + col] = acc[t][r];
        }
    }
}

// ---------------------------------------------------------------------------
// Self-loop contribution (also initializes accumulator over poisoned memory):
// acc[i, f] = dis[i]^2 * feat[i, f]     (f in [0,Fd), feat has row stride Fs)
// ---------------------------------------------------------------------------
__global__ void k_self_init(const float* __restrict__ feat, const float* __restrict__ dis,
                            float* __restrict__ acc, int n, int Fs, int Fd) {
    long long idx = (long long)blockIdx.x * blockDim.x + threadIdx.x;
    if (idx >= (long long)n * Fd) return;
    int i = (int)(idx / Fd);
    int f = (int)(idx - (long long)i * Fd);
    float d = dis[i];
    acc[idx] = d * d * feat[(long long)i * Fs + f];
}

// ---------------------------------------------------------------------------
// Edge scatter:  acc[dst, f] += dis[src]*ew*dis[dst] * feat[src, f]
// One thread per (edge, padded-feature). Scalar metadata is WGP$/L2 resident;
// the cost is the L2 atomic-add stream (L2-resident working set).
// ---------------------------------------------------------------------------
__global__ void k_edge_scatter(const int* __restrict__ src, const int* __restrict__ dst,
                               const float* __restrict__ ew, const float* __restrict__ dis,
                               const float* __restrict__ feat, float* __restrict__ acc,
                               long long total, int shift, int Fs, int Fd) {
    long long idx = (long long)blockIdx.x * blockDim.x + threadIdx.x;
    if (idx >= total) return;
    int e = (int)(idx >> shift);
    int f = (int)(idx & ((1 << shift) - 1));
    if (f >= Fd) return;
    int s = src[e];
    int d = dst[e];
    float nrm = dis[s] * ew[e] * dis[d];
    atomicAdd(&acc[(long long)d * Fd + f], nrm * feat[(long long)s * Fs + f]);
}

// ---------------------------------------------------------------------------
// Bias (+optional ReLU), in place.
// ---------------------------------------------------------------------------
__global__ void k_bias_act(float* __restrict__ a, const float* __restrict__ b,
                           int n, int F, int relu) {
    long long idx = (long long)blockIdx.x * blockDim.x + threadIdx.x;
    if (idx >= (long long)n * F) return;
    int f = (int)(idx % F);
    float v = a[idx] + b[f];
    a[idx] = relu ? fmaxf(v, 0.0f) : v;
}

// ---------------------------------------------------------------------------
// Launch
// ---------------------------------------------------------------------------
extern "C" void kernel_launch(void* const* d_in, const int* in_sizes, int n_in,
                              void* d_out, int out_size, void* d_ws, size_t ws_size,
                              hipStream_t stream) {
    (void)n_in; (void)out_size; (void)ws_size;

    const float* x  = (const float*)d_in[0];   // [N,128]
    const int*   ei = (const int*)  d_in[1];   // [2,E]
    const float* ew = (const float*)d_in[2];   // [E]
    const float* W1 = (const float*)d_in[3];   // [128,64]
    const float* b1 = (const float*)d_in[4];   // [64]
    const float* W2 = (const float*)d_in[5];   // [64,10]
    const float* b2 = (const float*)d_in[6];   // [10]

    const int CIN = 128, CH = 64, COUT = 10, CHPAD = 16;
    const int N = in_sizes[0] / CIN;
    const int E = in_sizes[2];
    const int* src = ei;
    const int* dst = ei + E;
    float* out = (float*)d_out;                // [N,10]

    // Workspace (floats): dis[N] | xw[N*64] | acc/h[N*64] | hw[N*16] |
    //                     w1p[(128/2)*64*2] | w2p[(64/2)*16*2]
    float* dis = (float*)d_ws;
    float* xw  = dis + N;
    float* acc = xw + (size_t)N * CH;
    float* hw  = acc + (size_t)N * CH;
    float* w1p = hw + (size_t)N * CHPAD;
    float* w2p = w1p + (size_t)(CIN / 2) * CH * 2;

    auto cdiv = [](long long a, long long b) { return (unsigned int)((a + b - 1) / b); };
    const int B256 = 256;

    // ---- normalization: dis = rsqrt(1 + segsum(ew @ dst)) ----
    k_deg_init<<<cdiv(N, B256), B256, 0, stream>>>(dis, N);
    k_deg_scatter<<<cdiv(E, B256), B256, 0, stream>>>(dst, ew, dis, E);
    k_rsqrt_inplace<<<cdiv(N, B256), B256, 0, stream>>>(dis, N);

    // ---- pack weights into WMMA-fragment layout (tiny, once) ----
    k_pack_b<<<cdiv((CIN / 2) * CH, B256), B256, 0, stream>>>(W1, w1p, CIN, CH, CH);
    k_pack_b<<<cdiv((CH / 2) * CHPAD, B256), B256, 0, stream>>>(W2, w2p, CH, COUT, CHPAD);

    // ---- layer 1 ----
    const int tiles = (N + 15) / 16;
    k_gemm_wmma_f32<128, 4><<<cdiv(tiles, 4), 128, 0, stream>>>(x, w1p, xw, N);
    k_self_init<<<cdiv((long long)N * CH, B256), B256, 0, stream>>>(xw, dis, acc, N, CH, CH);
    k_edge_scatter<<<cdiv((long long)E * 64, B256), B256, 0, stream>>>(
        src, dst, ew, dis, xw, acc, (long long)E * 64, 6, CH, CH);
    k_bias_act<<<cdiv((long long)N * CH, B256), B256, 0, stream>>>(acc, b1, N, CH, 1);

    // ---- layer 2 ----
    k_gemm_wmma_f32<64, 1><<<cdiv(tiles, 4), 128, 0, stream>>>(acc, w2p, hw, N);
    k_self_init<<<cdiv((long long)N * COUT, B256), B256, 0, stream>>>(hw, dis, out, N, CHPAD, COUT);
    k_edge_scatter<<<cdiv((long long)E * 16, B256), B256, 0, stream>>>(
        src, dst, ew, dis, hw, out, (long long)E * 16, 4, CHPAD, COUT);
    k_bias_act<<<cdiv((long long)N * COUT, B256), B256, 0, stream>>>(out, b2, N, COUT, 0);
}